// ARAPLoss_89661737271727
// MI455X (gfx1250) — compile-verified
//
#include <hip/hip_runtime.h>
#include <hip/hip_bf16.h>

typedef float v2f __attribute__((ext_vector_type(2)));
typedef float v8f __attribute__((ext_vector_type(8)));

// ---------------------------------------------------------------------------
// Wave32 sum reduction. Primary path: V_WMMA_F32_16X16X4_F32 with B = ones.
// A-matrix (16x4 f32) layout: lanes 0-15 hold rows m=lane (K=0 in reg0, K=1 in
// reg1); lanes 16-31 hold rows m=lane-16 (K=2,K=3).  With a = {s, 0} every
// lane's partial lands in exactly one A element; D = A x ones_B gives per-row
// sums replicated across all 16 columns.  Lane L holds rows (L<16 ? 0..7 :
// 8..15) in c[0..7], so summing c[0..7] and xor-shuffling across the half-wave
// boundary yields the full 32-lane total in every lane.
// ---------------------------------------------------------------------------
__device__ __forceinline__ float wave_sum32(float s) {
#if __has_builtin(__builtin_amdgcn_wmma_f32_16x16x4_f32)
    v2f a; a.x = s;   a.y = 0.0f;
    v2f b; b.x = 1.0f; b.y = 1.0f;
    v8f c = {};
    c = __builtin_amdgcn_wmma_f32_16x16x4_f32(
        /*neg_a=*/false, a, /*neg_b=*/false, b,
        /*c_mod=*/(short)0, c, /*reuse_a=*/false, /*reuse_b=*/false);
    float t = ((c[0] + c[1]) + (c[2] + c[3])) + ((c[4] + c[5]) + (c[6] + c[7]));
    t += __shfl_xor(t, 16, 32);
    return t;
#else
    #pragma unroll
    for (int off = 16; off > 0; off >>= 1) s += __shfl_xor(s, off, 32);
    return s;
#endif
}

__global__ void arap_init(double* ws) {
    if (blockIdx.x == 0 && threadIdx.x == 0) ws[0] = 0.0;
}

// One lane per (point, neighbor) pair.  idx/dist streams: fully coalesced,
// non-temporal (read once; keep L2 free for the 12 MB pc gather table).
__global__ __launch_bounds__(256) void arap_main(
    const float* __restrict__ pc,
    const long long* __restrict__ nn_idx,
    const float* __restrict__ nn_dist,
    double* __restrict__ ws,
    int num_pairs)
{
    int tid    = blockIdx.x * blockDim.x + threadIdx.x;
    int stride = gridDim.x * blockDim.x;

    float acc = 0.0f;
    for (int p = tid; p < num_pairs; p += stride) {
        int i = p >> 4;                                   // point id (K = 16)
        long long j64 = __builtin_nontemporal_load(&nn_idx[p]);
        float     dnn = __builtin_nontemporal_load(&nn_dist[p]);
        int j = (int)j64;

        // pc[i]: same address across each half-wave -> broadcast.
        float xi = pc[3 * i + 0];
        float yi = pc[3 * i + 1];
        float zi = pc[3 * i + 2];
        // pc[j]: random gather, L2-resident (12 MB table, 192 MB L2).
        float xj = pc[3 * j + 0];
        float yj = pc[3 * j + 1];
        float zj = pc[3 * j + 2];

        float dx = xi - xj, dy = yi - yj, dz = zi - zj;
        float sq = fmaf(dx, dx, fmaf(dy, dy, dz * dz));
        acc += fabsf(sq - dnn);
    }

    // All lanes reconverged here: EXEC is all-ones, legal for WMMA.
    float wsum = wave_sum32(acc);

    __shared__ float waveSums[8];
    int lane = threadIdx.x & 31;
    int wave = threadIdx.x >> 5;
    if (lane == 0) waveSums[wave] = wsum;
    __syncthreads();

    if (wave == 0) {
        float v = (lane < 8) ? waveSums[lane] : 0.0f;
        v += __shfl_xor(v, 4, 32);
        v += __shfl_xor(v, 2, 32);
        v += __shfl_xor(v, 1, 32);
        if (lane == 0) atomicAdd(ws, (double)v);          // global_atomic_add_f64
    }
}

__global__ void arap_final(const double* __restrict__ ws,
                           float* __restrict__ out, double inv_nk) {
    if (blockIdx.x == 0 && threadIdx.x == 0)
        out[0] = (float)(ws[0] * inv_nk);
}

extern "C" void kernel_launch(void* const* d_in, const int* in_sizes, int n_in,
                              void* d_out, int out_size, void* d_ws, size_t ws_size,
                              hipStream_t stream) {
    const float*     pc      = (const float*)d_in[0];      // (N, 3) f32
    const long long* nn_idx  = (const long long*)d_in[1];  // (N, 16) i64
    const float*     nn_dist = (const float*)d_in[2];      // (N, 16) f32
    float*           out     = (float*)d_out;
    double*          ws      = (double*)d_ws;

    const int num_pairs = in_sizes[1];                     // N * 16

    int blocks = (num_pairs + 255) / 256;
    if (blocks > 2048) blocks = 2048;                      // grid-stride covers rest
    if (blocks < 1)    blocks = 1;

    arap_init<<<1, 32, 0, stream>>>(ws);
    arap_main<<<blocks, 256, 0, stream>>>(pc, nn_idx, nn_dist, ws, num_pairs);
    arap_final<<<1, 32, 0, stream>>>(ws, out, 1.0 / (double)num_pairs);
}